// HGT_3977139716779
// MI455X (gfx1250) — compile-verified
//
#include <hip/hip_runtime.h>
#include <math.h>

typedef __attribute__((ext_vector_type(2))) float v2f;
typedef __attribute__((ext_vector_type(8))) float v8f;

#define HIDW 64

enum { EPI_PLAIN = 0, EPI_RELU = 1, EPI_SKIP = 2 };

__device__ __forceinline__ float gelu_exact(float x) {
    return 0.5f * x * (1.0f + erff(x * 0.7071067811865476f));
}

__device__ __forceinline__ void atomicMaxF(float* addr, float val) {
    // float atomic max via signed/unsigned int ordering trick
    if (val >= 0.0f) atomicMax((int*)addr, __float_as_int(val));
    else             atomicMin((unsigned int*)addr, __float_as_uint(val));
}

__global__ __launch_bounds__(256)
void fill_kernel(float* __restrict__ p, float v, int n) {
    int i = blockIdx.x * blockDim.x + threadIdx.x;
    if (i < n) p[i] = v;
}

// in-place exact GELU (keeps erff's data-dependent branches out of the GEMM)
__global__ __launch_bounds__(256)
void gelu_kernel(float* __restrict__ p, int n) {
    int i = blockIdx.x * blockDim.x + threadIdx.x;
    if (i < n) p[i] = gelu_exact(p[i]);
}

// D[N,64] = epi( A[N,F] @ B[F,64] + bias[64] ), per-wave 16x64 stripe.
// fp32 WMMA 16x16x4:
//  A frag (2 VGPR): lane r=L&15 -> row, hi=L>>4: holds K = kb+2*hi, kb+2*hi+1
//  B frag (2 VGPR): lane r -> col, holds rows kb+2*hi, kb+2*hi+1
//  C/D frag (8 VGPR): element (i + 8*hi, r) in register i
__global__ __launch_bounds__(256)
void gemm64_wmma_kernel(const float* __restrict__ A, int F,
                        const float* __restrict__ B,
                        const float* __restrict__ bias,
                        float* __restrict__ D,
                        const float* __restrict__ Hold,
                        const float* __restrict__ gatep,
                        int nrows, int epi)
{
    const int lane = threadIdx.x & 31;
    const int wave = threadIdx.x >> 5;
    const int tile = blockIdx.x * 8 + wave;         // 16-row tile
    if (tile * 16 >= nrows) return;                 // wave-uniform exit
    const int r  = lane & 15;
    const int hi = lane >> 4;
    const long rowbase = (long)(tile * 16 + r) * F;

    v8f acc0 = {}; v8f acc1 = {}; v8f acc2 = {}; v8f acc3 = {};

    const int Fa = F & ~3;                          // aligned portion
    #pragma unroll 2
    for (int kb = 0; kb < Fa; kb += 4) {
        const int k0 = kb + 2 * hi;
        const v2f av = *(const v2f*)(A + rowbase + k0); // b64 load, 8B aligned (F even)
        const float* B0 = B + (long)k0 * HIDW;
        const float* B1 = B0 + HIDW;
        v2f bv;
        bv.x = B0[r];      bv.y = B1[r];
        acc0 = __builtin_amdgcn_wmma_f32_16x16x4_f32(false, av, false, bv, (short)0, acc0, false, false);
        bv.x = B0[16 + r]; bv.y = B1[16 + r];
        acc1 = __builtin_amdgcn_wmma_f32_16x16x4_f32(false, av, false, bv, (short)0, acc1, false, false);
        bv.x = B0[32 + r]; bv.y = B1[32 + r];
        acc2 = __builtin_amdgcn_wmma_f32_16x16x4_f32(false, av, false, bv, (short)0, acc2, false, false);
        bv.x = B0[48 + r]; bv.y = B1[48 + r];
        acc3 = __builtin_amdgcn_wmma_f32_16x16x4_f32(false, av, false, bv, (short)0, acc3, false, false);
    }
    if (Fa < F) {
        // single tail iteration (F=334): clamp addresses, zero only the A frag.
        // A=0 annihilates the clamped (finite) B rows, so B needs no masking.
        const int k0  = Fa + 2 * hi;
        const int k0c = min(k0,     F - 1);
        const int k1c = min(k0 + 1, F - 1);
        float a0 = A[rowbase + k0c];
        float a1 = A[rowbase + k1c];
        a0 = (k0     < F) ? a0 : 0.0f;              // v_cndmask, no branch
        a1 = (k0 + 1 < F) ? a1 : 0.0f;
        v2f av; av.x = a0; av.y = a1;
        const float* B0 = B + (long)k0c * HIDW;
        const float* B1 = B + (long)k1c * HIDW;
        v2f bv;
        bv.x = B0[r];      bv.y = B1[r];
        acc0 = __builtin_amdgcn_wmma_f32_16x16x4_f32(false, av, false, bv, (short)0, acc0, false, false);
        bv.x = B0[16 + r]; bv.y = B1[16 + r];
        acc1 = __builtin_amdgcn_wmma_f32_16x16x4_f32(false, av, false, bv, (short)0, acc1, false, false);
        bv.x = B0[32 + r]; bv.y = B1[32 + r];
        acc2 = __builtin_amdgcn_wmma_f32_16x16x4_f32(false, av, false, bv, (short)0, acc2, false, false);
        bv.x = B0[48 + r]; bv.y = B1[48 + r];
        acc3 = __builtin_amdgcn_wmma_f32_16x16x4_f32(false, av, false, bv, (short)0, acc3, false, false);
    }

    float gate = 0.0f;
    if (epi == EPI_SKIP) gate = 1.0f / (1.0f + expf(-gatep[0]));

    #pragma unroll
    for (int c = 0; c < 4; ++c) {
        v8f acc = (c == 0) ? acc0 : (c == 1) ? acc1 : (c == 2) ? acc2 : acc3;
        const int col = c * 16 + r;
        const float bvv = bias[col];
        #pragma unroll
        for (int i = 0; i < 8; ++i) {
            const int m = tile * 16 + i + 8 * hi;
            float dv = acc[i] + bvv;
            if (epi == EPI_RELU) {
                dv = fmaxf(dv, 0.0f);
            } else if (epi == EPI_SKIP) {
                dv = gate * dv + (1.0f - gate) * Hold[(long)m * HIDW + col];
            }
            D[(long)m * HIDW + col] = dv;
        }
    }
}

// one thread per (edge, head):
// score = sum_d k[d] * (A_row_d . q) * P[h] * scale   (refactored einsum)
__global__ __launch_bounds__(256)
void score_kernel(const float* __restrict__ K, const float* __restrict__ Q,
                  const float* __restrict__ Ar, const float* __restrict__ Pr,
                  const int* __restrict__ src, const int* __restrict__ dst,
                  int E, float scale,
                  float* __restrict__ ex, float* __restrict__ nmax)
{
    int idx = blockIdx.x * blockDim.x + threadIdx.x;
    if (idx >= E * 8) return;
    const int e = idx >> 3, h = idx & 7;
    const int sn = src[e], dn = dst[e];
    const float* kp = K + (long)sn * 64 + h * 8;    // 32B aligned
    const float* qp = Q + (long)dn * 64 + h * 8;
    const float4 k0 = *(const float4*)(kp);
    const float4 k1 = *(const float4*)(kp + 4);
    const float4 q0 = *(const float4*)(qp);
    const float4 q1 = *(const float4*)(qp + 4);
    const float kr[8] = {k0.x, k0.y, k0.z, k0.w, k1.x, k1.y, k1.z, k1.w};
    const float* Am = Ar + h * 64;                  // [8,8] row-major (d,f)
    float sc = 0.0f;
    #pragma unroll
    for (int d2 = 0; d2 < 8; ++d2) {
        const float4 a0 = *(const float4*)(Am + d2 * 8);
        const float4 a1 = *(const float4*)(Am + d2 * 8 + 4);
        float dq = a0.x * q0.x + a0.y * q0.y + a0.z * q0.z + a0.w * q0.w
                 + a1.x * q1.x + a1.y * q1.y + a1.z * q1.z + a1.w * q1.w;
        sc = fmaf(kr[d2], dq, sc);
    }
    sc *= Pr[h] * scale;
    ex[idx] = sc;
    atomicMaxF(&nmax[dn * 8 + h], sc);
}

__global__ __launch_bounds__(256)
void expsum_kernel(float* __restrict__ ex, const float* __restrict__ nmax,
                   float* __restrict__ nsum, const int* __restrict__ dst, int E)
{
    int idx = blockIdx.x * blockDim.x + threadIdx.x;
    if (idx >= E * 8) return;
    const int e = idx >> 3, h = idx & 7;
    const int dn = dst[e];
    float v = expf(ex[idx] - nmax[dn * 8 + h]);
    ex[idx] = v;
    atomicAdd(&nsum[dn * 8 + h], v);
}

__global__ __launch_bounds__(256)
void msg_kernel(const float* __restrict__ V, const float* __restrict__ Mr,
                const float* __restrict__ ex, const float* __restrict__ nsum,
                const int* __restrict__ src, const int* __restrict__ dst,
                int E, float* __restrict__ out)
{
    int idx = blockIdx.x * blockDim.x + threadIdx.x;
    if (idx >= E * 8) return;
    const int e = idx >> 3, h = idx & 7;
    const int sn = src[e], dn = dst[e];
    const float alpha = ex[idx] / nsum[dn * 8 + h];
    const float* vp = V + (long)sn * 64 + h * 8;
    const float4 v0 = *(const float4*)(vp);
    const float4 v1 = *(const float4*)(vp + 4);
    const float vr[8] = {v0.x, v0.y, v0.z, v0.w, v1.x, v1.y, v1.z, v1.w};
    const float* Mm = Mr + h * 64;
    float acc[8] = {0, 0, 0, 0, 0, 0, 0, 0};
    #pragma unroll
    for (int d2 = 0; d2 < 8; ++d2) {
        const float4 m0 = *(const float4*)(Mm + d2 * 8);
        const float4 m1 = *(const float4*)(Mm + d2 * 8 + 4);
        const float vd = vr[d2];
        acc[0] = fmaf(vd, m0.x, acc[0]); acc[1] = fmaf(vd, m0.y, acc[1]);
        acc[2] = fmaf(vd, m0.z, acc[2]); acc[3] = fmaf(vd, m0.w, acc[3]);
        acc[4] = fmaf(vd, m1.x, acc[4]); acc[5] = fmaf(vd, m1.y, acc[5]);
        acc[6] = fmaf(vd, m1.z, acc[6]); acc[7] = fmaf(vd, m1.w, acc[7]);
    }
    float* ob = out + (long)dn * 64 + h * 8;
    #pragma unroll
    for (int f = 0; f < 8; ++f) atomicAdd(ob + f, alpha * acc[f]);
}

// one thread per row: all 4 logits, float4 loads of W_out rows
__global__ __launch_bounds__(256)
void outproj_kernel(const float* __restrict__ h, const float* __restrict__ W,
                    const float* __restrict__ b, float* __restrict__ out, int N)
{
    int row = blockIdx.x * blockDim.x + threadIdx.x;
    if (row >= N) return;
    const float* hr = h + (long)row * 64;
    float4 acc = *(const float4*)b;
    #pragma unroll
    for (int k4 = 0; k4 < 16; ++k4) {
        const float4 hv = *(const float4*)(hr + k4 * 4);
        const float4 w0 = *(const float4*)(W + (k4 * 4 + 0) * 4);
        const float4 w1 = *(const float4*)(W + (k4 * 4 + 1) * 4);
        const float4 w2 = *(const float4*)(W + (k4 * 4 + 2) * 4);
        const float4 w3 = *(const float4*)(W + (k4 * 4 + 3) * 4);
        acc.x = fmaf(hv.x, w0.x, acc.x); acc.y = fmaf(hv.x, w0.y, acc.y);
        acc.z = fmaf(hv.x, w0.z, acc.z); acc.w = fmaf(hv.x, w0.w, acc.w);
        acc.x = fmaf(hv.y, w1.x, acc.x); acc.y = fmaf(hv.y, w1.y, acc.y);
        acc.z = fmaf(hv.y, w1.z, acc.z); acc.w = fmaf(hv.y, w1.w, acc.w);
        acc.x = fmaf(hv.z, w2.x, acc.x); acc.y = fmaf(hv.z, w2.y, acc.y);
        acc.z = fmaf(hv.z, w2.z, acc.z); acc.w = fmaf(hv.z, w2.w, acc.w);
        acc.x = fmaf(hv.w, w3.x, acc.x); acc.y = fmaf(hv.w, w3.y, acc.y);
        acc.z = fmaf(hv.w, w3.z, acc.z); acc.w = fmaf(hv.w, w3.w, acc.w);
    }
    *(float4*)(out + (long)row * 4) = acc;
}

extern "C" void kernel_launch(void* const* d_in, const int* in_sizes, int n_in,
                              void* d_out, int out_size, void* d_ws, size_t ws_size,
                              hipStream_t stream)
{
    static const int Nn[4] = {100000, 200000, 20000, 2000};
    static const int Ff[4] = {334, 512, 128, 128};
    static const int Ee[6] = {800000, 800000, 1000000, 1000000, 200000, 200000};
    static const int RS[6] = {0, 1, 1, 2, 1, 3};
    static const int RD[6] = {1, 0, 2, 1, 3, 1};
    const float scale = 0.35355339059327373f; // 1/sqrt(8)

    const float* x[4];   for (int t = 0; t < 4; ++t) x[t]   = (const float*)d_in[t];
    const float* Win[4]; for (int t = 0; t < 4; ++t) Win[t] = (const float*)d_in[4 + 2 * t];
    const float* bin[4]; for (int t = 0; t < 4; ++t) bin[t] = (const float*)d_in[5 + 2 * t];
    const float* KW   = (const float*)d_in[12];
    const float* QW   = (const float*)d_in[13];
    const float* VW   = (const float*)d_in[14];
    const float* AW   = (const float*)d_in[15];
    const float* Kbv  = (const float*)d_in[16];
    const float* Qbv  = (const float*)d_in[17];
    const float* Vbv  = (const float*)d_in[18];
    const float* Abv  = (const float*)d_in[19];
    const float* skip = (const float*)d_in[20];
    const float* A_rel = (const float*)d_in[21];
    const float* M_rel = (const float*)d_in[22];
    const float* P_rel = (const float*)d_in[23];
    const float* W_out = (const float*)d_in[24];
    const float* b_out = (const float*)d_in[25];
    const int* ei[6]; for (int r = 0; r < 6; ++r) ei[r] = (const int*)d_in[26 + r];

    // workspace carve (floats)
    float* ws = (float*)d_ws;
    size_t off = 0;
    float *bufH[4], *bufK[4], *bufQ[4], *bufV[4], *bufO[4];
    for (int t = 0; t < 4; ++t) { bufH[t] = ws + off; off += (size_t)Nn[t] * 64; }
    for (int t = 0; t < 4; ++t) { bufK[t] = ws + off; off += (size_t)Nn[t] * 64; }
    for (int t = 0; t < 4; ++t) { bufQ[t] = ws + off; off += (size_t)Nn[t] * 64; }
    for (int t = 0; t < 4; ++t) { bufV[t] = ws + off; off += (size_t)Nn[t] * 64; }
    for (int t = 0; t < 4; ++t) { bufO[t] = ws + off; off += (size_t)Nn[t] * 64; }
    float* edge_ex = ws + off; off += (size_t)1000000 * 8;
    float* nmax    = ws + off; off += (size_t)200000 * 8;
    float* nsum    = ws + off; off += (size_t)200000 * 8;

    // input projections: h[t] = relu(x @ W_in + b_in)
    for (int t = 0; t < 4; ++t) {
        int tiles = Nn[t] / 16;
        gemm64_wmma_kernel<<<dim3((tiles + 7) / 8), 256, 0, stream>>>(
            x[t], Ff[t], Win[t], bin[t], bufH[t], nullptr, nullptr,
            Nn[t], EPI_RELU);
    }

    for (int l = 0; l < 2; ++l) {
        // K/Q/V projections + zero aggregation buffers
        for (int t = 0; t < 4; ++t) {
            int tiles = Nn[t] / 16;
            dim3 g((tiles + 7) / 8);
            const size_t wo = (size_t)(l * 4 + t) * 64 * 64;
            const size_t bo = (size_t)(l * 4 + t) * 64;
            gemm64_wmma_kernel<<<g, 256, 0, stream>>>(bufH[t], 64, KW + wo, Kbv + bo,
                bufK[t], nullptr, nullptr, Nn[t], EPI_PLAIN);
            gemm64_wmma_kernel<<<g, 256, 0, stream>>>(bufH[t], 64, QW + wo, Qbv + bo,
                bufQ[t], nullptr, nullptr, Nn[t], EPI_PLAIN);
            gemm64_wmma_kernel<<<g, 256, 0, stream>>>(bufH[t], 64, VW + wo, Vbv + bo,
                bufV[t], nullptr, nullptr, Nn[t], EPI_PLAIN);
            int nel = Nn[t] * 64;
            fill_kernel<<<(nel + 255) / 256, 256, 0, stream>>>(bufO[t], 0.0f, nel);
        }
        // relations: segment softmax + scatter
        for (int r = 0; r < 6; ++r) {
            const int s = RS[r], d = RD[r], E = Ee[r];
            const float* Ar = A_rel + (size_t)(l * 6 + r) * 8 * 64;
            const float* Mr = M_rel + (size_t)(l * 6 + r) * 8 * 64;
            const float* Pr = P_rel + (size_t)(l * 6 + r) * 8;
            const int* srcI = ei[r];
            const int* dstI = ei[r] + E;
            const int nd8 = Nn[d] * 8;
            fill_kernel<<<(nd8 + 255) / 256, 256, 0, stream>>>(nmax, -3.0e38f, nd8);
            fill_kernel<<<(nd8 + 255) / 256, 256, 0, stream>>>(nsum, 0.0f, nd8);
            const int nthr = E * 8;
            dim3 ge((nthr + 255) / 256);
            score_kernel<<<ge, 256, 0, stream>>>(bufK[s], bufQ[d], Ar, Pr, srcI, dstI,
                                                 E, scale, edge_ex, nmax);
            expsum_kernel<<<ge, 256, 0, stream>>>(edge_ex, nmax, nsum, dstI, E);
            msg_kernel<<<ge, 256, 0, stream>>>(bufV[s], Mr, edge_ex, nsum, srcI, dstI,
                                               E, bufO[d]);
        }
        // node update: h = g*(gelu(out)@AW + Ab) + (1-g)*h
        // GELU applied in-place on bufO first (keeps GEMM loop branch-free)
        for (int t = 0; t < 4; ++t) {
            int tiles = Nn[t] / 16;
            const size_t wo = (size_t)(l * 4 + t) * 64 * 64;
            const size_t bo = (size_t)(l * 4 + t) * 64;
            int nel = Nn[t] * 64;
            gelu_kernel<<<(nel + 255) / 256, 256, 0, stream>>>(bufO[t], nel);
            gemm64_wmma_kernel<<<dim3((tiles + 7) / 8), 256, 0, stream>>>(
                bufO[t], 64, AW + wo, Abv + bo, bufH[t], bufH[t],
                skip + (size_t)l * 4 + t, Nn[t], EPI_SKIP);
        }
    }

    // final: author logits = h[0] @ W_out + b_out
    {
        outproj_kernel<<<(100000 + 255) / 256, 256, 0, stream>>>(
            bufH[0], W_out, b_out, (float*)d_out, 100000);
    }
}